// TimeFrequencyRepresentation_70884140253755
// MI455X (gfx1250) — compile-verified
//
#include <hip/hip_runtime.h>
#include <hip/hip_bf16.h>
#include <math.h>

// ---------------- problem constants ----------------
#define NBC   2048      // 32*64 signals
#define NT    1024      // samples per signal
#define NSC   27        // scales (widths 4..30)
#define PAD   256       // LDS zero-pad on each side of the signal

// Scale groups for 16-row WMMA tiles (K-steps padded to multiples of 4)
#define KSTEP0 80       // scales 4..19 : u in [-160, 159]  (support +-(8*19+1))
#define UMIN0  (-160)
#define KSTEP1 124      // scales 20..30: u in [-248, 247]  (support +-(8*30+1))
#define UMIN1  (-248)
#define AFRAG_TOTAL ((KSTEP0 + KSTEP1) * 32)   // float4 entries

typedef float v2f __attribute__((ext_vector_type(2)));
typedef float v8f __attribute__((ext_vector_type(8)));

// ---------------- setup 1: int_psi (inclusive fp64 scan) ----------------
__global__ __launch_bounds__(1024) void cwt_int_psi(double2* __restrict__ ip) {
  __shared__ double2 buf[1024];
  const int i = threadIdx.x;
  const double x = -8.0 + 16.0 * (double)i / 1023.0;
  const double env = exp(-x * x) * 0.5641895835477563;  // pi^-1/2 (Bw=1)
  const double ang = 2.0 * M_PI * x;                    // Cf=1
  buf[i].x = env * cos(ang);
  buf[i].y = env * sin(ang);
  __syncthreads();
  for (int off = 1; off < 1024; off <<= 1) {
    double2 add;
    add.x = (i >= off) ? buf[i - off].x : 0.0;
    add.y = (i >= off) ? buf[i - off].y : 0.0;
    __syncthreads();
    buf[i].x += add.x;
    buf[i].y += add.y;
    __syncthreads();
  }
  const double step = 16.0 / 1023.0;
  double2 r; r.x = buf[i].x * step; r.y = buf[i].y * step;
  ip[i] = r;   // cumsum(psi)*step (pywt's conj cancels under abs())
}

// ---------------- setup 2: bake A-fragments in WMMA lane layout ----------------
__device__ inline double2 cwt_fval(const double2* ip, int s, int m) {
  // filter[m] = int_psi[ j_{16s-m} ] for m in [0,16s] (reversed resample), else 0
  double2 z; z.x = 0.0; z.y = 0.0;
  if (m < 0 || m > 16 * s) return z;
  const long long k = (long long)(16 * s - m);
  const int j = (int)((k * 1023LL) / (long long)(16 * s));
  return ip[j];
}
__device__ inline double2 cwt_hval(const double2* ip, int s, int u) {
  // h_s[u] = g_s[8s-u], g_s[m] = f[m] - f[m-1], support m in [0, 16s+1]
  double2 z; z.x = 0.0; z.y = 0.0;
  const int m = 8 * s - u;
  if (m < 0 || m > 16 * s + 1) return z;
  const double2 a = cwt_fval(ip, s, m);
  const double2 b = cwt_fval(ip, s, m - 1);
  double2 r; r.x = a.x - b.x; r.y = a.y - b.y;
  return r;
}

__global__ __launch_bounds__(256) void cwt_build_afrag(const double2* __restrict__ ip,
                                                       float4* __restrict__ af) {
  const int idx = blockIdx.x * 256 + threadIdx.x;
  if (idx >= AFRAG_TOTAL) return;
  const int g     = (idx >= KSTEP0 * 32) ? 1 : 0;
  const int local = idx - (g ? KSTEP0 * 32 : 0);
  const int k     = local >> 5;
  const int lane  = local & 31;
  const int row   = lane & 15;          // A-matrix M = row  (lanes 0-15 == 16-31)
  const int khalf = (lane >> 4) * 2;    // lanes 16-31 hold K=2,3
  const int wi    = g * 16 + row;       // width index 0..26 (else zero-pad row)
  float4 r = make_float4(0.f, 0.f, 0.f, 0.f);
  if (wi < NSC) {
    const int s    = 4 + wi;
    const int umin = g ? UMIN1 : UMIN0;
    const int u0   = umin + 4 * k + khalf;
    const double2 h0 = cwt_hval(ip, s, u0);
    const double2 h1 = cwt_hval(ip, s, u0 + 1);
    r = make_float4((float)h0.x, (float)h1.x, (float)h0.y, (float)h1.y);
  }
  af[idx] = r;
}

// ---------------- main: WMMA FIR bank ----------------
// One scale-group, two 16-wide time tiles per call. KSTEPS/UMIN/G are
// compile-time so the K-loop is a uniform SGPR-counted loop (EXEC stays all-1s).
template <int KSTEPS, int UMIN, int G>
__device__ __forceinline__ void cwt_tilepair(const float* __restrict__ sx,
                                             const float4* __restrict__ afrag,
                                             float* __restrict__ out,
                                             int bc, int lane, int t0a, int t0b) {
  const int row   = lane & 15;           // B column (== output t offset)
  const int khalf = (lane >> 4) * 2;     // second half-wave -> K=2,3
  const float4* __restrict__ af = afrag + (G ? (KSTEP0 * 32) : 0);

  v8f aR0 = {}, aI0 = {}, aR1 = {}, aI1 = {};
  const int b0 = PAD + t0a + row + UMIN + khalf;
  const int b1 = PAD + t0b + row + UMIN + khalf;

#pragma unroll 2
  for (int k = 0; k < KSTEPS; ++k) {
    const float4 a = af[k * 32 + lane];          // 512B/wave, L2-resident stream
    v2f ar; ar[0] = a.x; ar[1] = a.y;            // Re(h) taps
    v2f ai; ai[0] = a.z; ai[1] = a.w;            // Im(h) taps
    v2f bb0; bb0[0] = sx[b0 + 4 * k]; bb0[1] = sx[b0 + 4 * k + 1];
    v2f bb1; bb1[0] = sx[b1 + 4 * k]; bb1[1] = sx[b1 + 4 * k + 1];
    aR0 = __builtin_amdgcn_wmma_f32_16x16x4_f32(false, ar, false, bb0,
                                                (short)0, aR0, false, false);
    aI0 = __builtin_amdgcn_wmma_f32_16x16x4_f32(false, ai, false, bb0,
                                                (short)0, aI0, false, false);
    aR1 = __builtin_amdgcn_wmma_f32_16x16x4_f32(false, ar, false, bb1,
                                                (short)0, aR1, false, false);
    aI1 = __builtin_amdgcn_wmma_f32_16x16x4_f32(false, ai, false, bb1,
                                                (short)0, aI1, false, false);
  }

  // C/D layout: VGPR r -> M = r (lanes 0-15) / r+8 (lanes 16-31), N = lane%16
  const int moff = (lane >> 4) << 3;
#pragma unroll
  for (int r = 0; r < 8; ++r) {
    const int wi = G * 16 + r + moff;
    if (wi < NSC) {
      const float sc = (float)(4 + wi);
      const float m0 = sqrtf((aR0[r] * aR0[r] + aI0[r] * aI0[r]) * sc);
      const float m1 = sqrtf((aR1[r] * aR1[r] + aI1[r] * aI1[r]) * sc);
      float* op = out + ((size_t)bc * NSC + wi) * NT + row;
      op[t0a] = m0;
      op[t0b] = m1;
    }
  }
}

__global__ __launch_bounds__(256) void cwt_wmma_main(const float* __restrict__ x,
                                                     const float4* __restrict__ afrag,
                                                     float* __restrict__ out) {
  __shared__ float sx[NT + 2 * PAD];
  const int bc = blockIdx.x;                      // signal index (b*64+c)
  const float* xin = x + (size_t)bc * NT;
  for (int i = threadIdx.x; i < NT + 2 * PAD; i += 256) {
    const int src = i - PAD;
    sx[i] = (src >= 0 && src < NT) ? xin[src] : 0.0f;
  }
  __syncthreads();

  const int wave = threadIdx.x >> 5;              // 0..7 (wave32)
  const int lane = threadIdx.x & 31;

  // 64 time tiles; each pass covers 16 tiles (8 waves x 2 tiles), 4 passes.
  for (int pass = 0; pass < 4; ++pass) {
    const int t0a = ((pass << 4) + wave) << 4;    // tiles pass*16 + wave
    const int t0b = t0a + 128;                    // ... and + 8 tiles
    cwt_tilepair<KSTEP0, UMIN0, 0>(sx, afrag, out, bc, lane, t0a, t0b);
    cwt_tilepair<KSTEP1, UMIN1, 1>(sx, afrag, out, bc, lane, t0a, t0b);
  }
}

// ---------------- launcher ----------------
extern "C" void kernel_launch(void* const* d_in, const int* in_sizes, int n_in,
                              void* d_out, int out_size, void* d_ws, size_t ws_size,
                              hipStream_t stream) {
  const float* x = (const float*)d_in[0];
  float* out = (float*)d_out;

  // workspace layout: [0, 16KB) int_psi (double2 x1024), then A-fragment stream
  double2* ip = (double2*)d_ws;
  float4* af  = (float4*)((char*)d_ws + 1024 * sizeof(double2));

  cwt_int_psi<<<1, 1024, 0, stream>>>(ip);
  cwt_build_afrag<<<(AFRAG_TOTAL + 255) / 256, 256, 0, stream>>>(ip, af);
  cwt_wmma_main<<<NBC, 256, 0, stream>>>(x, af, out);
}